// glp_rotation_pool_7146825581229
// MI455X (gfx1250) — compile-verified
//
#include <hip/hip_runtime.h>

#define HH 128
#define WW 128
#define AA 24
#define GG 6   // A / kernel_size

typedef float v2f __attribute__((ext_vector_type(2)));

// One thread per (b,c,h,w): gathers the 3 rotated taps per g via the gfx1250
// async-to-LDS pipe (ASYNCcnt), identity tap via direct loads, max-reduces,
// writes 6 contiguous floats with non-temporal stores.
__global__ __launch_bounds__(256) void glp_rotation_pool_kernel(
    const float* __restrict__ img, float* __restrict__ out) {
  // Per-lane 96B staging region: slot (lane, g, k) at byte lane*96 + g*16 + k*4
  __shared__ float stage[256 * 24];

  const int tid = threadIdx.x;
  const int w  = tid & (WW - 1);
  const int h  = ((blockIdx.x & 63) << 1) | (tid >> 7);
  const int bc = blockIdx.x >> 6;                 // 0..127, consecutive blocks share a plane

  const float* plane = img + (size_t)bc * (HH * WW * AA);

  // LDS byte offset of this lane's staging base (low 32 bits of generic ptr)
  const unsigned lds_base = (unsigned)(size_t)(&stage[tid * 24]);

  const float fx = (float)w - 63.5f;
  const float fy = (float)h - 63.5f;

  // cos/sin of f32(deg2rad(-15k)), k=1..3, correctly rounded to f32.
  const float cs[3] = { 0.96592581272125244f, 0.86602541804313660f, 0.70710676908493042f };
  const float sn[3] = { -0.25881904363632202f, -0.5f, -0.70710676908493042f };

  bool valid[3];
#pragma unroll
  for (int j = 0; j < 3; ++j) {
    // Match XLA semantics exactly: separate rounding for each mul/add, no FMA.
#pragma clang fp contract(off)
    const float c = cs[j], s = sn[j];
    // x_src = c*xg + s*yg + cx ; y_src = -s*xg + c*yg + cy
    const float xs = (c * fx + s * fy) + 63.5f;
    const float ys = ((-s) * fx + c * fy) + 63.5f;
    int xi = (int)rintf(xs);   // round half to even, matches jnp.round
    int yi = (int)rintf(ys);
    valid[j] = (xi >= 0) & (xi < WW) & (yi >= 0) & (yi < HH);
    xi = min(max(xi, 0), WW - 1);
    yi = min(max(yi, 0), HH - 1);

    // byte offset of element (yi, xi, a=k) inside the plane; a=4g+k adds g*16
    const unsigned voff  = (unsigned)(yi * WW + xi) * (AA * 4) + (unsigned)(j + 1) * 4;
    const unsigned laddr = lds_base + (unsigned)(j + 1) * 4;
    // 6 async per-lane gathers; the same immediate offset (g*16) applies to
    // both the global address and the LDS destination (ISA §4.4).
    asm volatile(
        "global_load_async_to_lds_b32 %0, %1, %2 offset:0\n\t"
        "global_load_async_to_lds_b32 %0, %1, %2 offset:16\n\t"
        "global_load_async_to_lds_b32 %0, %1, %2 offset:32\n\t"
        "global_load_async_to_lds_b32 %0, %1, %2 offset:48\n\t"
        "global_load_async_to_lds_b32 %0, %1, %2 offset:64\n\t"
        "global_load_async_to_lds_b32 %0, %1, %2 offset:80"
        :: "v"(laddr), "v"(voff), "s"(plane) : "memory");
  }

  // k = 0 is the exact identity rotation: direct loads of a = 4g.
  const float* p0 = plane + (size_t)(h * WW + w) * AA;
  float r[GG];
#pragma unroll
  for (int g = 0; g < GG; ++g) r[g] = p0[g * 4];

  // Wait for this wave's async gathers to land in LDS.
  asm volatile("s_wait_asynccnt 0" ::: "memory");

#pragma unroll
  for (int j = 0; j < 3; ++j) {
#pragma unroll
    for (int g = 0; g < GG; ++g) {
      float v = stage[tid * 24 + g * 4 + (j + 1)];
      v = valid[j] ? v : 0.0f;   // zero fill outside the image
      r[g] = fmaxf(r[g], v);
    }
  }

  // 24B per thread, contiguous across the wave; NT stores keep L2 for gathers.
  float* op = out + ((size_t)bc * (HH * WW) + (size_t)(h * WW + w)) * GG;
  v2f v01 = {r[0], r[1]};
  v2f v23 = {r[2], r[3]};
  v2f v45 = {r[4], r[5]};
  __builtin_nontemporal_store(v01, (v2f*)(op + 0));
  __builtin_nontemporal_store(v23, (v2f*)(op + 2));
  __builtin_nontemporal_store(v45, (v2f*)(op + 4));
}

extern "C" void kernel_launch(void* const* d_in, const int* in_sizes, int n_in,
                              void* d_out, int out_size, void* d_ws, size_t ws_size,
                              hipStream_t stream) {
  (void)in_sizes; (void)n_in; (void)d_ws; (void)ws_size; (void)out_size;
  const float* img = (const float*)d_in[0];   // (8,16,128,128,24) f32
  float* out = (float*)d_out;                 // (8,16,128,128,6)  f32
  // 8*16 planes * 64 row-pairs = 8192 blocks of 256 threads (8 wave32)
  dim3 grid(8 * 16 * (HH / 2));
  dim3 block(256);
  glp_rotation_pool_kernel<<<grid, block, 0, stream>>>(img, out);
}